// EnhancedUltra_71227737637030
// MI455X (gfx1250) — compile-verified
//
#include <hip/hip_runtime.h>
#include <math.h>

typedef __attribute__((ext_vector_type(2))) float v2f;
typedef __attribute__((ext_vector_type(8))) float v8f;

#define BQ 64
#define RN 256
#define DD 64

// ---------------- kernel 0: zero workspace (counts + freq) ----------------
__global__ void zero_ws_kernel(float* counts, unsigned* freq) {
    int i = blockIdx.x * blockDim.x + threadIdx.x;
    if (i < BQ * RN) counts[i] = 0.0f;
    if (i < RN) freq[i] = 0u;
}

// ---------------- kernel 1: stream edges -------------------------------
// - LDS 256-bin histogram for relation frequency
// - LDS 32768-bit Bloom filter over query entity ids (hash = id & 32767)
//   -> almost every edge skips the 64-way exact-match loop
__global__ __launch_bounds__(256) void edge_scan_kernel(
    const int* __restrict__ edge_index,      // [2, E] : src then dst
    const int* __restrict__ edge_type,       // [E]
    const int* __restrict__ query_entities,  // [BQ]
    int E,
    float* __restrict__ counts,              // [BQ, RN]
    unsigned* __restrict__ freq)             // [RN]
{
    __shared__ unsigned lfreq[RN];
    __shared__ unsigned bloom[1024];         // 32768 bits

    const int tid = threadIdx.x;
    for (int i = tid; i < RN; i += 256) lfreq[i] = 0u;
    for (int i = tid; i < 1024; i += 256) bloom[i] = 0u;
    __syncthreads();
    if (tid < BQ) {
        unsigned h = ((unsigned)query_entities[tid]) & 32767u;
        atomicOr(&bloom[h >> 5], 1u << (h & 31u));
    }
    __syncthreads();

    const int* __restrict__ src = edge_index;
    const int* __restrict__ dst = edge_index + E;
    const int stride = gridDim.x * blockDim.x;

    for (int e = blockIdx.x * blockDim.x + tid; e < E; e += stride) {
        const int s = src[e];
        const int d = dst[e];
        const int t = edge_type[e];
        atomicAdd(&lfreq[t], 1u);

        const unsigned hs = ((unsigned)s) & 32767u;
        const unsigned hd = ((unsigned)d) & 32767u;
        const unsigned hit = ((bloom[hs >> 5] >> (hs & 31u)) |
                              (bloom[hd >> 5] >> (hd & 31u))) & 1u;
        if (hit) {  // rare path (~0.5% incl. false positives)
            for (int b = 0; b < BQ; ++b) {
                const int q = query_entities[b];
                if (s == q || d == q) atomicAdd(&counts[b * RN + t], 1.0f);
            }
        }
    }
    __syncthreads();
    if (lfreq[tid]) atomicAdd(&freq[tid], lfreq[tid]);
}

// ---------------- WMMA f32 16x16x4 tile GEMM helper ----------------------
// Computes Y[M0:M0+16, N0:N0+16] = relu( X[M0:M0+16, 0:K] @ W[0:K, N0:N0+16] + bias )
// Operand placement per CDNA5 ISA 7.12.2:
//   A 16x4 : lanes 0-15 row M=l, VGPR0/1 = K0,K1 ; lanes 16-31 = K2,K3
//   B 4x16 : lanes 0-15 col N=l, VGPR0/1 = K0,K1 ; lanes 16-31 = K2,K3
//   C/D    : VGPR v, lanes 0-15 -> (M=v, N=l); lanes 16-31 -> (M=v+8, N=l)
__device__ inline void wmma_tile_relu(const float* __restrict__ X, int ldx,
                                      const float* __restrict__ W, int ldw,
                                      const float* __restrict__ bias,
                                      float* __restrict__ Y, int ldy,
                                      int M0, int N0, int K, int lane)
{
    const int half = lane >> 4;
    const int l    = lane & 15;
    v8f acc = {};
    for (int k0 = 0; k0 < K; k0 += 4) {
        v2f a, b;
        const float* xr = X + (M0 + l) * ldx + (k0 + 2 * half);
        a.x = xr[0];
        a.y = xr[1];
        const float* wr = W + (k0 + 2 * half) * ldw + (N0 + l);
        b.x = wr[0];
        b.y = wr[ldw];
        acc = __builtin_amdgcn_wmma_f32_16x16x4_f32(
            false, a, false, b, (short)0, acc, false, false);
    }
    const float bv = bias[N0 + l];
#pragma unroll
    for (int v = 0; v < 8; ++v) {
        float val = acc[v] + bv;
        Y[(M0 + v + 8 * half) * ldy + (N0 + l)] = val > 0.0f ? val : 0.0f;
    }
}

// ---------------- kernel 2: finalize features + WMMA MLP ------------------
__global__ __launch_bounds__(512) void head_kernel(
    const float* __restrict__ emb,          // [BQ, RN, DD]
    const float* __restrict__ W1, const float* __restrict__ b1,   // [132,64],[64]
    const float* __restrict__ W2, const float* __restrict__ b2,   // [64,32],[32]
    const float* __restrict__ W3, const float* __restrict__ b3,   // [32,16],[16]
    const float* __restrict__ W4, const float* __restrict__ b4,   // [16,1],[1]
    const int* __restrict__ query_rels,     // [BQ]
    const int* __restrict__ num_nodes_p,    // [1]
    const float* __restrict__ counts,       // [BQ, RN]
    const unsigned* __restrict__ freq,      // [RN]
    float* __restrict__ out, int E)
{
    __shared__ float feats[BQ][132];  // [rel_emb(64) | entity_emb(64) | stats(4)]
    __shared__ float h1[BQ][64];
    __shared__ float h2[BQ][32];
    __shared__ float g[BQ][16];
    __shared__ float sdeg[BQ];

    const int tid = threadIdx.x;

    // degree per query
    if (tid < BQ) {
        float s = 0.0f;
        const float* cb = counts + tid * RN;
        for (int r = 0; r < RN; ++r) s += cb[r];
        sdeg[tid] = s;
    }
    __syncthreads();

    // entity_emb (counts . emb / deg) and rel_emb; coalesced over d
    {
        const int bg = tid >> 6;     // 0..7
        const int d  = tid & 63;
        for (int b = bg; b < BQ; b += 8) {
            const float* cb = counts + b * RN;
            const float* eb = emb + (size_t)(b * RN) * DD + d;
            float acc = 0.0f;
            for (int r = 0; r < RN; ++r) acc += cb[r] * eb[(size_t)r * DD];
            const float dg = sdeg[b];
            feats[b][64 + d] = dg > 0.0f ? acc / dg : 0.0f;
            feats[b][d] = emb[((size_t)b * RN + query_rels[b]) * DD + d];
        }
    }
    // stats
    if (tid < BQ) {
        const int b = tid;
        const float invE = 1.0f / (float)E;
        const float rfn = fminf((float)freq[query_rels[b]] * invE, 1.0f);
        const float edn = fminf(sdeg[b] * invE, 1.0f);
        const float nn  = (float)num_nodes_p[0];
        const float dens = fminf((float)E / fmaxf(nn * nn, 1.0f), 1.0f);
        feats[b][128] = rfn;
        feats[b][129] = edn;
        feats[b][130] = rfn;
        feats[b][131] = dens;
    }
    __syncthreads();

    const int wave = tid >> 5;
    const int lane = tid & 31;

    // layer 1: [64,132]@[132,64] -> 4x4 tiles of 16x16, K=132 (33 wmma steps)
    if (wave < 16)
        wmma_tile_relu(&feats[0][0], 132, W1, 64, b1, &h1[0][0], 64,
                       (wave >> 2) * 16, (wave & 3) * 16, 132, lane);
    __syncthreads();

    // layer 2: [64,64]@[64,32] -> 4x2 tiles, K=64
    if (wave < 8)
        wmma_tile_relu(&h1[0][0], 64, W2, 32, b2, &h2[0][0], 32,
                       (wave >> 1) * 16, (wave & 1) * 16, 64, lane);
    __syncthreads();

    // layer 3: [64,32]@[32,16] -> 4x1 tiles, K=32
    if (wave < 4)
        wmma_tile_relu(&h2[0][0], 32, W3, 16, b3, &g[0][0], 16,
                       wave * 16, 0, 32, lane);
    __syncthreads();

    // layer 4 + sigmoid
    if (tid < BQ) {
        float acc = b4[0];
#pragma unroll
        for (int k = 0; k < 16; ++k) acc += g[tid][k] * W4[k];
        out[tid] = 1.0f / (1.0f + expf(-acc));
    }
}

// ---------------- launcher ------------------------------------------------
extern "C" void kernel_launch(void* const* d_in, const int* in_sizes, int n_in,
                              void* d_out, int out_size, void* d_ws, size_t ws_size,
                              hipStream_t stream) {
    const float* emb = (const float*)d_in[0];
    const float* W1  = (const float*)d_in[1];
    const float* b1  = (const float*)d_in[2];
    const float* W2  = (const float*)d_in[3];
    const float* b2  = (const float*)d_in[4];
    const float* W3  = (const float*)d_in[5];
    const float* b3  = (const float*)d_in[6];
    const float* W4  = (const float*)d_in[7];
    const float* b4  = (const float*)d_in[8];
    const int* query_rels     = (const int*)d_in[9];
    const int* query_entities = (const int*)d_in[10];
    const int* edge_index     = (const int*)d_in[11];
    const int* edge_type      = (const int*)d_in[12];
    const int* num_nodes      = (const int*)d_in[13];
    const int E = in_sizes[12];

    float* counts  = (float*)d_ws;
    unsigned* freq = (unsigned*)((char*)d_ws + (size_t)BQ * RN * sizeof(float));

    zero_ws_kernel<<<(BQ * RN + 255) / 256, 256, 0, stream>>>(counts, freq);
    edge_scan_kernel<<<512, 256, 0, stream>>>(edge_index, edge_type,
                                              query_entities, E, counts, freq);
    head_kernel<<<1, 512, 0, stream>>>(emb, W1, b1, W2, b2, W3, b3, W4, b4,
                                       query_rels, num_nodes, counts, freq,
                                       (float*)d_out, E);
}